// Filter_24885040513526
// MI455X (gfx1250) — compile-verified
//
#include <hip/hip_runtime.h>
#include <hip/hip_bf16.h>

typedef __attribute__((ext_vector_type(2))) float v2f;
typedef __attribute__((ext_vector_type(8))) float v8f;

#define LEN   2048
#define SZ    17
#define PADW  8
#define KP    20          // K padded 17 -> 20 (5 x WMMA K=4 steps)
#define NCH   2

// One workgroup (8 waves) per (16-row l-tile, (b,c) pair).
// out[bc][l][m] = sum_t G[l][t] * xrow[m + t],  G = window(x) @ filt[c], K padded to 20.
__global__ __launch_bounds__(256) void Filter_24885040513526_kernel(
    const float* __restrict__ x,     // [8, 2, 2048]
    const float* __restrict__ filt,  // [1, 2, 17, 17]
    float* __restrict__ out)         // [8, 2, 2048, 2048]
{
    __shared__ float xrow[LEN + 2 * PADW + 4];   // padded signal row (zeros outside), 2068
    __shared__ float fsh[SZ * SZ];               // filter for this channel
    __shared__ float g[16][KP];                  // G tile, K zero-padded to 20

    const int l0  = blockIdx.x * 16;
    const int bc  = blockIdx.y;        // b*2 + c
    const int c   = bc & 1;
    const int tid = threadIdx.x;

    const float* xsrc = x + (size_t)bc * LEN;

    // Stage 0: signal row (with zero halo) + filter into LDS.
    for (int i = tid; i < LEN + 2 * PADW + 4; i += 256) {
        int pos = i - PADW;
        xrow[i] = (pos >= 0 && pos < LEN) ? xsrc[pos] : 0.0f;
    }
    for (int i = tid; i < SZ * SZ; i += 256) fsh[i] = filt[c * SZ * SZ + i];
    __syncthreads();

    // Stage 1: G[row][t] = sum_s xrow[l0+row+s] * F[s][t]; t >= 17 -> 0 (kills K-pad garbage in B).
    for (int e = tid; e < 16 * KP; e += 256) {
        int row = e / KP, t = e % KP;
        float acc = 0.0f;
        if (t < SZ) {
            #pragma unroll
            for (int s = 0; s < SZ; ++s) acc += xrow[l0 + row + s] * fsh[s * SZ + t];
        }
        g[row][t] = acc;
    }
    __syncthreads();

    // Stage 2: WMMA sweep over m.
    const int wave = tid >> 5;
    const int lane = tid & 31;
    const int half = lane >> 4;   // 0: lanes 0-15, 1: lanes 16-31
    const int ln16 = lane & 15;
    const int koff = half * 2;    // A/B K sub-offset per ISA 16x4 / 4x16 layout

    // A fragments (16x4 f32 per step): lane row = ln16, VGPR0 -> K=koff, VGPR1 -> K=koff+1.
    v2f a[5];
    #pragma unroll
    for (int k = 0; k < 5; ++k) {
        a[k].x = g[ln16][4 * k + koff];
        a[k].y = g[ln16][4 * k + koff + 1];
    }

    float* obase = out + (size_t)bc * LEN * LEN + (size_t)l0 * LEN;

    for (int j = 0; j < 16; ++j) {
        const int m0 = (wave + 8 * j) * 16;
        const int m  = m0 + ln16;

        // B fragments (4x16 f32 per step): lane col = ln16, VGPR0 -> K=koff, VGPR1 -> K=koff+1.
        v2f b[5];
        #pragma unroll
        for (int k = 0; k < 5; ++k) {
            b[k].x = xrow[m + 4 * k + koff];
            b[k].y = xrow[m + 4 * k + koff + 1];
        }

        v8f acc = {};
        #pragma unroll
        for (int k = 0; k < 5; ++k) {
            acc = __builtin_amdgcn_wmma_f32_16x16x4_f32(
                /*neg_a=*/false, a[k], /*neg_b=*/false, b[k],
                /*c_mod=*/(short)0, acc, /*reuse_a=*/false, /*reuse_b=*/false);
        }

        // D layout: VGPR v -> row (v + 8*half), col ln16. 8 coalesced b32 stores.
        #pragma unroll
        for (int v = 0; v < 8; ++v) {
            const int lrow = v + 8 * half;
            obase[(size_t)lrow * LEN + m] = acc[v];
        }
    }
}

extern "C" void kernel_launch(void* const* d_in, const int* in_sizes, int n_in,
                              void* d_out, int out_size, void* d_ws, size_t ws_size,
                              hipStream_t stream) {
    const float* x    = (const float*)d_in[0];   // [8,2,2048] f32
    const float* filt = (const float*)d_in[1];   // [1,2,17,17] f32
    float* out        = (float*)d_out;           // [8,2,2048,2048] f32

    dim3 grid(LEN / 16, 8 * NCH);   // (128, 16)
    Filter_24885040513526_kernel<<<grid, 256, 0, stream>>>(x, filt, out);
}